// AKOrNBlock_10213432230577
// MI455X (gfx1250) — compile-verified
//
#include <hip/hip_runtime.h>
#include <hip/hip_bf16.h>

typedef __attribute__((ext_vector_type(2))) float v2f;
typedef __attribute__((ext_vector_type(8))) float v8f;

#define N_OSC 256
#define DIM 64
#define LSTR 65            // padded LDS row stride (floats) -> conflict-free rows
#define OSTR 65            // Omega LDS stride
#define NSTEPS 10
#define EPSN 1e-12f

__launch_bounds__(256)
__global__ void akorn_kernel(const float* __restrict__ x_in,
                             const float* __restrict__ etap,
                             const float* __restrict__ W,      // (N,N)
                             const float* __restrict__ A,      // (d,d)
                             const float* __restrict__ h,      // (N,d)
                             const float* __restrict__ w_ro,   // (d,1)
                             const float* __restrict__ b_ro,   // (1,)
                             float* __restrict__ out,          // (B,1)
                             float* __restrict__ x_out)        // (B,N,d)
{
    __shared__ float xs [N_OSC * LSTR];   // current oscillator state x_b
    __shared__ float drv[N_OSC * LSTR];   // drive = h + W@x + x@Omega
    __shared__ float omg[DIM * OSTR];     // Omega = A - A^T (block-resident)

    const int b    = blockIdx.x;
    const int t    = threadIdx.x;      // 0..255
    const int lane = t & 31;
    const int wave = t >> 5;           // 0..7
    const int lrow = lane & 15;        // M/N index within tile
    const int lhi  = lane >> 4;        // K-pair / M+8 selector
    const float eta = etap[0];

    const float* xb = x_in + (size_t)b * N_OSC * DIM;

    // ---- stage x_b into LDS (coalesced) + build Omega ----
    for (int idx = t; idx < N_OSC * DIM; idx += 256)
        xs[(idx >> 6) * LSTR + (idx & 63)] = xb[idx];
    for (int idx = t; idx < DIM * DIM; idx += 256) {
        const int c = idx >> 6, e = idx & 63;
        omg[c * OSTR + e] = A[c * DIM + e] - A[e * DIM + c];
    }
    __syncthreads();

    // ---- initial per-row L2 normalize (thread t owns row t) ----
    {
        float s = 0.f;
        #pragma unroll
        for (int e = 0; e < DIM; ++e) { float v = xs[t*LSTR + e]; s += v * v; }
        float inv = 1.f / fmaxf(sqrtf(s), EPSN);
        #pragma unroll
        for (int e = 0; e < DIM; ++e) xs[t*LSTR + e] *= inv;
    }
    __syncthreads();

    const int tm0   = wave;            // first row-tile (rows tm0*16 ..)
    const int tm1   = wave + 8;        // second row-tile
    const int arow0 = tm0 * 16 + lrow;
    const int arow1 = tm1 * 16 + lrow;

    for (int step = 0; step < NSTEPS; ++step) {
        // ===== GEMM phase: drv = h + W @ x + x @ Omega (V_WMMA_F32_16X16X4_F32)
        // Each wave owns 2 row-tiles x 4 col-tiles; B fragments shared by both.
        {
            v8f acc[2][4];
            // init accumulators with h in C/D lane layout (coalesced 64B segs)
            #pragma unroll
            for (int ti = 0; ti < 2; ++ti) {
                const int rbase = (ti ? tm1 : tm0) * 16 + 8 * lhi;
                #pragma unroll
                for (int tn = 0; tn < 4; ++tn)
                    #pragma unroll
                    for (int r = 0; r < 8; ++r)
                        acc[ti][tn][r] = h[(rbase + r) * DIM + tn * 16 + lrow];
            }

            // interaction: K over 256 oscillators; A = W rows, B = xs
            for (int j0 = 0; j0 < N_OSC; j0 += 4) {
                const int ka = j0 + 2 * lhi;
                v2f a0, a1;
                a0.x = W[arow0 * N_OSC + ka]; a0.y = W[arow0 * N_OSC + ka + 1];
                a1.x = W[arow1 * N_OSC + ka]; a1.y = W[arow1 * N_OSC + ka + 1];
                #pragma unroll
                for (int tn = 0; tn < 4; ++tn) {
                    const int col = tn * 16 + lrow;
                    v2f bf; bf.x = xs[(ka    ) * LSTR + col];
                            bf.y = xs[(ka + 1) * LSTR + col];
                    acc[0][tn] = __builtin_amdgcn_wmma_f32_16x16x4_f32(
                        false, a0, false, bf, (short)0, acc[0][tn], false, false);
                    acc[1][tn] = __builtin_amdgcn_wmma_f32_16x16x4_f32(
                        false, a1, false, bf, (short)0, acc[1][tn], false, false);
                }
            }
            // natural rotation: K over 64; A = xs rows, B = Omega (LDS)
            for (int k0 = 0; k0 < DIM; k0 += 4) {
                const int kk = k0 + 2 * lhi;
                v2f a0, a1;
                a0.x = xs[arow0 * LSTR + kk]; a0.y = xs[arow0 * LSTR + kk + 1];
                a1.x = xs[arow1 * LSTR + kk]; a1.y = xs[arow1 * LSTR + kk + 1];
                #pragma unroll
                for (int tn = 0; tn < 4; ++tn) {
                    const int col = tn * 16 + lrow;
                    v2f bf; bf.x = omg[(kk    ) * OSTR + col];
                            bf.y = omg[(kk + 1) * OSTR + col];
                    acc[0][tn] = __builtin_amdgcn_wmma_f32_16x16x4_f32(
                        false, a0, false, bf, (short)0, acc[0][tn], false, false);
                    acc[1][tn] = __builtin_amdgcn_wmma_f32_16x16x4_f32(
                        false, a1, false, bf, (short)0, acc[1][tn], false, false);
                }
            }
            // spill D tiles to LDS (C/D layout: M = r + 8*lhi, N = lrow)
            #pragma unroll
            for (int ti = 0; ti < 2; ++ti) {
                const int rbase = (ti ? tm1 : tm0) * 16 + 8 * lhi;
                #pragma unroll
                for (int tn = 0; tn < 4; ++tn)
                    #pragma unroll
                    for (int r = 0; r < 8; ++r)
                        drv[(rbase + r) * LSTR + tn * 16 + lrow] = acc[ti][tn][r];
            }
        }
        __syncthreads();

        // ===== elementwise tangent update + renorm: thread t owns row t =====
        {
            const int n = t;
            float dot = 0.f;
            #pragma unroll
            for (int e = 0; e < DIM; ++e)
                dot += xs[n*LSTR + e] * drv[n*LSTR + e];
            float nrm2 = 0.f;
            #pragma unroll
            for (int e = 0; e < DIM; ++e) {
                float xv = xs[n*LSTR + e];
                float xn = xv + eta * (drv[n*LSTR + e] - dot * xv);
                drv[n*LSTR + e] = xn;            // stash un-normalized update
                nrm2 += xn * xn;
            }
            float inv = 1.f / fmaxf(sqrtf(nrm2), EPSN);
            #pragma unroll
            for (int e = 0; e < DIM; ++e)
                xs[n*LSTR + e] = drv[n*LSTR + e] * inv;
        }
        __syncthreads();
    }

    // ---- write final x (coalesced) ----
    float* xo = x_out + (size_t)b * N_OSC * DIM;
    for (int idx = t; idx < N_OSC * DIM; idx += 256)
        xo[idx] = xs[(idx >> 6) * LSTR + (idx & 63)];

    // ---- readout: out[b] = mean_n(x[n,:] . w_ro) + b_ro ----
    float ro = 0.f;
    #pragma unroll
    for (int e = 0; e < DIM; ++e) ro += xs[t*LSTR + e] * w_ro[e];
    __syncthreads();
    drv[t] = ro;
    __syncthreads();
    for (int s = 128; s > 0; s >>= 1) {
        if (t < s) drv[t] += drv[t + s];
        __syncthreads();
    }
    if (t == 0) out[b] = drv[0] * (1.f / (float)N_OSC) + b_ro[0];
}

extern "C" void kernel_launch(void* const* d_in, const int* in_sizes, int n_in,
                              void* d_out, int out_size, void* d_ws, size_t ws_size,
                              hipStream_t stream) {
    (void)n_in; (void)d_ws; (void)ws_size; (void)out_size;
    const float* x    = (const float*)d_in[0];
    const float* eta  = (const float*)d_in[1];
    const float* W    = (const float*)d_in[2];
    const float* A    = (const float*)d_in[3];
    const float* h    = (const float*)d_in[4];
    const float* w_ro = (const float*)d_in[5];
    const float* b_ro = (const float*)d_in[6];

    const int B = in_sizes[0] / (N_OSC * DIM);   // 1024
    float* out   = (float*)d_out;                // (B,1) first
    float* x_out = out + B;                      // then (B,N,d)

    akorn_kernel<<<B, 256, 0, stream>>>(x, eta, W, A, h, w_ro, b_ro, out, x_out);
}